// Conv_14654428414364
// MI455X (gfx1250) — compile-verified
//
#include <hip/hip_runtime.h>

#define N_NODES 65536
#define N_EDGES 1048576
#define D_FEAT  64

typedef __attribute__((ext_vector_type(2))) float v2f;
typedef __attribute__((ext_vector_type(8))) float v8f;

// ---------------------------------------------------------------------------
// Zero-init the output (harness poisons d_out with 0xAA before timing).
// ---------------------------------------------------------------------------
__global__ __launch_bounds__(256) void zero_f4_kernel(float4* __restrict__ out, int n4) {
    int i = blockIdx.x * blockDim.x + threadIdx.x;
    if (i < n4) out[i] = float4{0.f, 0.f, 0.f, 0.f};
}

// ---------------------------------------------------------------------------
// COO SpMM: one wave32 per 16-edge tile.
//   Per tile:  OUT_partial(16x64) = diag(val[16]) x X_gathered(16x64)
//   computed as 4 feature tiles of 16 cols, each a K-chain of four
//   V_WMMA_F32_16X16X4_F32 ops, then scattered with fp32 atomics.
//
// fp32 WMMA layouts (CDNA5 ISA 7.12.2, wave32):
//   A (16x4):  lanes 0-15 M=0..15 hold {K0,K1} in {v0,v1};
//              lanes 16-31 M=0..15 hold {K2,K3}.
//   B (4x16):  lanes 0-15 N=0..15 hold rows {K0,K1}; lanes 16-31 rows {K2,K3}.
//   D (16x16): VGPR r = row r (lanes 0-15) and row r+8 (lanes 16-31), N=lane%16.
// ---------------------------------------------------------------------------
__global__ __launch_bounds__(256) void spmm_coo_wmma_kernel(
    const float* __restrict__ x,
    const int*   __restrict__ row,
    const int*   __restrict__ col,
    const float* __restrict__ val,
    float*       __restrict__ out) {

    const int lane = threadIdx.x & 31;
    const int wave = blockIdx.x * (blockDim.x >> 5) + (threadIdx.x >> 5);
    const int e0   = wave << 4;        // first of this wave's 16 edges
    const int half = lane >> 4;        // 0: lanes 0-15, 1: lanes 16-31
    const int ln   = lane & 15;

    // Each lane caches metadata of edge (e0 + ln); lanes 16-31 mirror 0-15,
    // so wave shuffles from lanes 0..15 fetch any edge's (row, col, val).
    const int   eMine = e0 + ln;
    const float vMine = val[eMine];
    const int   cMine = col[eMine];
    const int   rMine = row[eMine];

    // Build the four K-chain A tiles of diag(val): A_j[m][k] = val[4j+k] iff m==4j+k.
    v2f a[4];
    int c0[4], c1[4];                  // gathered source rows per chain step
#pragma unroll
    for (int j = 0; j < 4; ++j) {
        const int m0 = 4 * j + 2 * half;      // K slot 0 of this lane-half
        const int m1 = m0 + 1;                // K slot 1 of this lane-half
        const float v0 = __shfl(vMine, m0, 32);
        const float v1 = __shfl(vMine, m1, 32);
        a[j].x = (ln == m0) ? v0 : 0.0f;
        a[j].y = (ln == m1) ? v1 : 0.0f;
        c0[j] = __shfl(cMine, m0, 32);
        c1[j] = __shfl(cMine, m1, 32);
    }

    // Scatter targets: D-matrix VGPR r holds rows r (half 0) / r+8 (half 1).
    int rT[8];
#pragma unroll
    for (int r = 0; r < 8; ++r)
        rT[r] = __shfl(rMine, r + 8 * half, 32);

#pragma unroll
    for (int t = 0; t < 4; ++t) {      // feature tile: columns 16t .. 16t+15
        v8f acc = {0.f, 0.f, 0.f, 0.f, 0.f, 0.f, 0.f, 0.f};
#pragma unroll
        for (int j = 0; j < 4; ++j) {  // K-chain over edges 4j..4j+3
            // B_j (4x16): row K = x[col[4j+K]][16t + N]. Each 16-lane half
            // loads 64 contiguous bytes of an x row -> coalesced, L2-resident.
            v2f b;
            b.x = x[c0[j] * D_FEAT + t * 16 + ln];
            b.y = x[c1[j] * D_FEAT + t * 16 + ln];
            acc = __builtin_amdgcn_wmma_f32_16x16x4_f32(
                /*neg_a=*/false, a[j], /*neg_b=*/false, b,
                /*c_mod=*/(short)0, acc, /*reuse_a=*/false, /*reuse_b=*/false);
        }
        // Scatter: D[m][n] = val[m] * x[col[m]][16t+n]  ->  out[row[m]][16t+n]
#pragma unroll
        for (int r = 0; r < 8; ++r) {
            atomicAdd(&out[rT[r] * D_FEAT + t * 16 + ln], acc[r]);
        }
    }
}

// ---------------------------------------------------------------------------
// Inputs (setup_inputs order): x[f32 N*D], row[i32 E], col[i32 E], val[f32 E],
// idx[i32 1] (no-op constant). Output: f32 N*D.
// ---------------------------------------------------------------------------
extern "C" void kernel_launch(void* const* d_in, const int* in_sizes, int n_in,
                              void* d_out, int out_size, void* d_ws, size_t ws_size,
                              hipStream_t stream) {
    const float* x   = (const float*)d_in[0];
    const int*   row = (const int*)  d_in[1];
    const int*   col = (const int*)  d_in[2];
    const float* val = (const float*)d_in[3];
    float*       out = (float*)d_out;

    (void)in_sizes; (void)n_in; (void)d_ws; (void)ws_size; (void)out_size;

    // Zero the accumulation target.
    const int n4 = (N_NODES * D_FEAT) / 4;            // 1,048,576 float4
    zero_f4_kernel<<<n4 / 256, 256, 0, stream>>>((float4*)out, n4);

    // One wave32 per 16 edges: 65536 waves -> 8192 blocks of 8 waves.
    const int waves  = N_EDGES / 16;
    const int blocks = waves / 8;
    spmm_coo_wmma_kernel<<<blocks, 256, 0, stream>>>(x, row, col, val, out);
}